// sparseKernelFT_20143396618627
// MI455X (gfx1250) — compile-verified
//
#include <hip/hip_runtime.h>
#include <stdint.h>

typedef float v2f __attribute__((ext_vector_type(2)));
typedef float v8f __attribute__((ext_vector_type(8)));

#define WMMA_F32(a, b, c) \
  __builtin_amdgcn_wmma_f32_16x16x4_f32(false, (a), false, (b), (short)0, (c), false, false)

#define AS1 __attribute__((address_space(1)))
#define AS3 __attribute__((address_space(3)))

#if __has_builtin(__builtin_amdgcn_global_load_async_to_lds_b32) && \
    __has_builtin(__builtin_amdgcn_s_wait_asynccnt)
#define HAVE_ASYNC_LDS 1
#else
#define HAVE_ASYNC_LDS 0
#endif

// ---------------- workspace layout (float offsets) ----------------
// twiddle tables
constexpr int OFF_WA = 0;      // [64][32]  t-forward (12 re | 12 im | pad)
constexpr int OFF_WB = 2048;   // [128][48] y/x-forward (Re/Im stacked K)
constexpr int OFF_WI = 8192;   // [48][128] x/y-inverse
constexpr int OFF_W3 = 14336;  // [24][64]  t-inverse (Hermitian, 1/N^3 folded)
constexpr int OFF_WF = 15872;  // [16][16]  lo_w transposed
// intermediate buffers (liveness-packed; peak ~185 MB)
constexpr size_t O_XA = 16384;     // 12582912 floats  [b,ch,ix][k3][c][iy]
constexpr size_t O_XB = 12599296;  //  4718592 floats  [b,ch][ky][k3][c][ix]
constexpr size_t O_XC = 17317888;  //  1769472 floats  [kx,ky,k3][b][c][i]
constexpr size_t O_XD = 16384;     //  1769472 floats  [b,o][ky][k3][c][kx]
constexpr size_t O_XE = 12599296;  //  4718592 floats  [b,o][ix][k3][c][ky]
constexpr size_t O_XF = 16384;     // 12582912 floats  [b,o][ix][iy][c][k3]
constexpr size_t O_YT = 12599296;  // 33554432 floats  [b][ix][iy][t][c]  (relu'd)

// ---------------- twiddle init (recomputed every call; deterministic) ----------------
__global__ __launch_bounds__(256) void init_tw(float* __restrict__ ws,
                                               const float* __restrict__ lo_w) {
  int g = blockIdx.x * 256 + threadIdx.x;
  const float c64 = 0.09817477042468103f;  // 2*pi/64
  if (g < 2048) {  // W_A: forward DFT over t, e^{-i th}
    int t = g >> 5, j = g & 31;
    float v = 0.f;
    if (j < 12)       v =  cosf(c64 * (float)((j * t) & 63));
    else if (j < 24)  v = -sinf(c64 * (float)(((j - 12) * t) & 63));
    ws[OFF_WA + g] = v;
  } else if (g < 8192) {  // W_B: forward over y/x, K-stacked Re/Im
    int i = g - 2048;
    int kk = i / 48, j = i - kk * 48;
    int iy = kk & 63;  bool imIn = kk >= 64;
    int kyI = j % 24;  bool imOut = j >= 24;
    int ky = kyI < 12 ? kyI : kyI + 40;  // active bins {0..11, 52..63}
    float th = c64 * (float)((ky * iy) & 63);
    float cv = cosf(th), sv = sinf(th);
    // Re2 = sum Re*cos + Im*sin ; Im2 = sum Im*cos - Re*sin
    ws[OFF_WB + i] = imIn ? (imOut ? cv : sv) : (imOut ? -sv : cv);
  } else if (g < 14336) {  // W_I: inverse over x/y, e^{+i th}
    int i = g - 8192;
    int r = i >> 7, cidx = i & 127;
    bool imIn = r >= 24; int kI = r % 24; int k = kI < 12 ? kI : kI + 40;
    bool imOut = cidx >= 64; int ix = cidx & 63;
    float th = c64 * (float)((k * ix) & 63);
    float cv = cosf(th), sv = sinf(th);
    // Re = R*cos - I*sin ; Im = R*sin + I*cos
    ws[OFF_WI + i] = imIn ? (imOut ? cv : -sv) : (imOut ? sv : cv);
  } else if (g < 15872) {  // W_I3: irfft over t (Hermitian weights, 1/64^3)
    int i = g - 14336;
    int r = i >> 6, t = i & 63;
    bool imIn = r >= 12; int k = imIn ? r - 12 : r;
    float s = (k == 0 ? 1.f : 2.f) * (1.f / 262144.f);
    float th = c64 * (float)((k * t) & 63);
    ws[OFF_W3 + i] = imIn ? -s * sinf(th) : s * cosf(th);
  } else if (g < 16128) {  // W_F[k=c][n=oc] = lo_w[oc][c]
    int i = g - 15872;
    ws[OFF_WF + i] = lo_w[(i & 15) * 16 + (i >> 4)];
  } else if (g < 16384) {
    ws[g] = 0.f;
  }
}

// ---------------- stage traits ----------------
struct StA {  // x[524288][64] -> XA ; cols: 12 re + 12 im (pad to 32)
  static constexpr int R = 524288, K = 64, NPAD = 32;
  static __device__ float post(float v, int, const float*) { return v; }
  static __device__ void store(float* D, unsigned row, int col, float v) {
    if (col >= 24) return;
    int comp = col >= 12;
    int k3 = comp ? col - 12 : col;
    D[(size_t)(row >> 6) * 1536 + (size_t)k3 * 128 + comp * 64 + (row & 63)] = v;
  }
};
struct StB {  // XA[98304][128] -> XB ; cols: 24 re + 24 im over ky
  static constexpr int R = 98304, K = 128, NPAD = 48;
  static __device__ float post(float v, int, const float*) { return v; }
  static __device__ void store(float* D, unsigned row, int col, float v) {
    unsigned k3 = row % 12u, ix = (row / 12u) & 63u, bch = row / 768u;
    int comp = col >= 24; int ky = comp ? col - 24 : col;
    D[((size_t)(bch * 24u + ky) * 12u + k3) * 128u + comp * 64 + ix] = v;
  }
};
struct StC {  // XB[36864][128] -> XC (per-mode [b][c][i] blocks)
  static constexpr int R = 36864, K = 128, NPAD = 48;
  static __device__ float post(float v, int, const float*) { return v; }
  static __device__ void store(float* D, unsigned row, int col, float v) {
    unsigned k3 = row % 12u, ky = (row / 12u) % 24u, bch = row / 288u;
    unsigned b = bch >> 4, ch = bch & 15u;
    int comp = col >= 24; int kx = comp ? col - 24 : col;
    D[((size_t)(kx * 24 + ky) * 12 + k3) * 256 + b * 32 + comp * 16 + ch] = v;
  }
};
struct I1 {  // XD[36864][48] -> XE ; cols: 64 re + 64 im over ix
  static constexpr int R = 36864, K = 48, NPAD = 128;
  static __device__ float post(float v, int, const float*) { return v; }
  static __device__ void store(float* D, unsigned row, int col, float v) {
    unsigned k3 = row % 12u, ky = (row / 12u) % 24u, bo = row / 288u;
    int comp = col >= 64; int ix = col & 63;
    D[((size_t)(bo * 64 + ix) * 12 + k3) * 48 + comp * 24 + ky] = v;
  }
};
struct I2 {  // XE[98304][48] -> XF ; cols: 64 re + 64 im over iy
  static constexpr int R = 98304, K = 48, NPAD = 128;
  static __device__ float post(float v, int, const float*) { return v; }
  static __device__ void store(float* D, unsigned row, int col, float v) {
    unsigned k3 = row % 12u, ix = (row / 12u) & 63u, bo = row / 768u;
    int comp = col >= 64; int iy = col & 63;
    D[((size_t)(bo * 64 + ix) * 64 + iy) * 24 + comp * 12 + k3] = v;
  }
};
struct I3 {  // XF[524288][24] -> YT (irfft over t + ReLU, transposed for conv)
  static constexpr int R = 524288, K = 24, NPAD = 64;
  static __device__ float post(float v, int, const float*) { return fmaxf(v, 0.f); }
  static __device__ void store(float* D, unsigned row, int col, float v) {
    unsigned iy = row & 63u, ix = (row >> 6) & 63u, o = (row >> 12) & 15u, b = row >> 16;
    D[(((size_t)(b * 64 + ix) * 64 + iy) * 64 + col) * 16 + o] = v;
  }
};
struct FIN {  // YT[2097152][16] x lo_w^T + bias -> out[b][oc][sp]
  static constexpr int R = 2097152, K = 16, NPAD = 16;
  static __device__ float post(float v, int col, const float* bias) { return v + bias[col]; }
  static __device__ void store(float* D, unsigned row, int col, float v) {
    unsigned b = row >> 18, sp = row & 262143u;
    D[(size_t)(b * 16 + col) * 262144 + sp] = v;
  }
};

// ---------------- generic WMMA GEMM: D = A[R][K] * B[K][NPAD], scatter stores ----------------
template <class TR>
__global__ __launch_bounds__(256) void wgemm(const float* __restrict__ A,
                                             const float* __restrict__ B,
                                             float* __restrict__ D,
                                             const float* __restrict__ bias) {
  constexpr int Kd = TR::K, NP = TR::NPAD, NT = NP / 16, NCH = Kd / 4;
  __shared__ __align__(16) float Blds[NP * Kd];  // transposed: Blds[n][k]
#if HAVE_ASYNC_LDS
  // CDNA5 async global->LDS staging of the twiddle tile (ASYNCcnt-tracked).
  for (int idx = threadIdx.x; idx < Kd * NP; idx += 256) {
    int k = idx / NP, n = idx - k * NP;
    __builtin_amdgcn_global_load_async_to_lds_b32(
        (AS1 int*)(uintptr_t)(B + idx),
        (AS3 int*)(uintptr_t)&Blds[n * Kd + k], 0, 0);
  }
  __builtin_amdgcn_s_wait_asynccnt(0);
  __syncthreads();
#else
  for (int idx = threadIdx.x; idx < Kd * NP; idx += 256) {
    int k = idx / NP, n = idx - k * NP;
    Blds[n * Kd + k] = B[idx];
  }
  __syncthreads();
#endif

  int lane = threadIdx.x & 31;
  unsigned wave = blockIdx.x * 8u + (threadIdx.x >> 5);
  unsigned r0 = wave * 16u;
  if (r0 >= (unsigned)TR::R) return;
  int m = lane & 15, half = lane >> 4, kb = half * 2;

  const float* arow = A + (size_t)(r0 + m) * Kd;
  __builtin_prefetch(arow, 0, 3);  // global_prefetch_b8

  v8f acc[NT] = {};
  for (int c = 0; c < NCH; ++c) {
    int kk = c * 4 + kb;  // A lane holds (kk, kk+1); B lane holds rows (kk, kk+1), col m
    v2f a = *(const v2f*)(arow + kk);
#pragma unroll
    for (int t = 0; t < NT; ++t) {
      v2f b = *(const v2f*)&Blds[(t * 16 + m) * Kd + kk];
      acc[t] = WMMA_F32(a, b, acc[t]);
    }
  }
#pragma unroll
  for (int t = 0; t < NT; ++t)
#pragma unroll
    for (int j = 0; j < 8; ++j) {
      unsigned row = r0 + j + 8 * half;  // D VGPR j: M=j (lanes 0-15), M=j+8 (lanes 16-31)
      int col = t * 16 + m;
      TR::store(D, row, col, TR::post(acc[t][j], col, bias));
    }
}

// ---------------- per-mode 16x16 complex channel mix (one wave per mode) ----------------
__global__ __launch_bounds__(256) void modal_mix(const float* __restrict__ XC,
                                                 const float* __restrict__ w1,
                                                 const float* __restrict__ w2,
                                                 const float* __restrict__ w3,
                                                 const float* __restrict__ w4,
                                                 float* __restrict__ XD) {
  int lane = threadIdx.x & 31;
  int mode = blockIdx.x * 8 + (threadIdx.x >> 5);  // 6912 modes total
  int k3 = mode % 12;
  int ky = (mode / 12) % 24;
  int kx = mode / 288;
  const float* w = (kx >= 12) ? (ky >= 12 ? w4 : w2) : (ky >= 12 ? w3 : w1);
  int mx = kx >= 12 ? kx - 12 : kx;
  int my = ky >= 12 ? ky - 12 : ky;
  size_t wmode = ((size_t)mx * 12 + my) * 12 + k3;  // offset within [i][o] block of 1728

  int m = lane & 15, half = lane >> 4, kb = half * 2, n = m;
  // Branchless padded-row handling: rows 8..15 read row (m&7) and are masked to 0,
  // so no EXEC manipulation happens near the WMMA chain.
  const float amsk = (m < 8) ? 1.f : 0.f;
  const float* Ab = XC + (size_t)mode * 256 + (size_t)(m & 7) * 32;  // row: [Ar(16)|Ai(16)]

  v8f aR = {}, aI = {};
  for (int c = 0; c < 8; ++c) {  // K = 32
    int kk = c * 4 + kb;
    v2f a = *(const v2f*)(Ab + kk);
    a.x *= amsk;
    a.y *= amsk;
    v2f bR, bI;
    {
      int kr = kk, i = kr & 15;
      size_t off = ((size_t)(i * 16 + n) * 1728 + wmode) * 2;
      float wr = w[off], wi = w[off + 1];
      bR.x = (kr >= 16) ? -wi : wr;  // [Wr ; -Wi]
      bI.x = (kr >= 16) ?  wr : wi;  // [Wi ;  Wr]
    }
    {
      int kr = kk + 1, i = kr & 15;
      size_t off = ((size_t)(i * 16 + n) * 1728 + wmode) * 2;
      float wr = w[off], wi = w[off + 1];
      bR.y = (kr >= 16) ? -wi : wr;
      bI.y = (kr >= 16) ?  wr : wi;
    }
    aR = WMMA_F32(a, bR, aR);
    aI = WMMA_F32(a, bI, aI);
  }
#pragma unroll
  for (int j = 0; j < 8; ++j) {
    int b = j + 8 * half;
    if (b < 8) {
      size_t base = (((size_t)(b * 16 + n) * 24 + ky) * 12 + k3) * 48 + kx;
      XD[base] = aR[j];        // comp = re
      XD[base + 24] = aI[j];   // comp = im
    }
  }
}

// ---------------- launch ----------------
extern "C" void kernel_launch(void* const* d_in, const int* in_sizes, int n_in,
                              void* d_out, int out_size, void* d_ws, size_t ws_size,
                              hipStream_t stream) {
  (void)in_sizes; (void)n_in; (void)out_size; (void)ws_size;
  const float* x    = (const float*)d_in[0];
  const float* w1   = (const float*)d_in[1];
  const float* w2   = (const float*)d_in[2];
  const float* w3   = (const float*)d_in[3];
  const float* w4   = (const float*)d_in[4];
  const float* lo_w = (const float*)d_in[5];
  const float* lo_b = (const float*)d_in[6];
  float* ws  = (float*)d_ws;
  float* out = (float*)d_out;

  float* XA = ws + O_XA; float* XB = ws + O_XB; float* XC = ws + O_XC;
  float* XD = ws + O_XD; float* XE = ws + O_XE; float* XF = ws + O_XF;
  float* YT = ws + O_YT;

  init_tw<<<64, 256, 0, stream>>>(ws, lo_w);
  wgemm<StA><<<StA::R / 128, 256, 0, stream>>>(x,  ws + OFF_WA, XA, nullptr);
  wgemm<StB><<<StB::R / 128, 256, 0, stream>>>(XA, ws + OFF_WB, XB, nullptr);
  wgemm<StC><<<StC::R / 128, 256, 0, stream>>>(XB, ws + OFF_WB, XC, nullptr);
  modal_mix<<<864, 256, 0, stream>>>(XC, w1, w2, w3, w4, XD);
  wgemm<I1><<<I1::R / 128, 256, 0, stream>>>(XD, ws + OFF_WI, XE, nullptr);
  wgemm<I2><<<I2::R / 128, 256, 0, stream>>>(XE, ws + OFF_WI, XF, nullptr);
  wgemm<I3><<<I3::R / 128, 256, 0, stream>>>(XF, ws + OFF_W3, YT, nullptr);
  wgemm<FIN><<<FIN::R / 128, 256, 0, stream>>>(YT, ws + OFF_WF, out, lo_b);
}